// GaussianFiler_61040075211219
// MI455X (gfx1250) — compile-verified
//
#include <hip/hip_runtime.h>
#include <hip/hip_bf16.h>

// GaussianFiler fused kernel for MI455X (gfx1250, wave32).
// Phase 1: per-(b,f) block computes desc[r,k] = S1/(S0+eps) into LDS
//          (S0/S1 are V-contractions of rho/theta Gaussians), ~180k exps/block.
// Phase 2: 5 waves each run a 16x16 WMMA tile: desc(16x80) @ W(80x16-col-tile)
//          using V_WMMA_F32_16X16X4_F32 chained over K (20 steps), then
//          max over the 16 rotation rows + bias + relu, store to out.

typedef float v2f __attribute__((ext_vector_type(2)));
typedef float v8f __attribute__((ext_vector_type(8)));

#define B_      1024
#define V_      128
#define F_      5
#define BINS_   80
#define NROT_   16
#define DSTRIDE 81            // LDS row stride (odd -> bank-conflict free)
#define TWO_PI      6.283185307179586f
#define INV_TWO_PI  0.15915494309189535f
#define ROT_STEP    (TWO_PI / (float)NROT_)
#define EPS_        1e-5f

__global__ __launch_bounds__(160)
void GaussianFiler_kernel(const float* __restrict__ feature,
                          const float* __restrict__ rhos,
                          const float* __restrict__ thetas,
                          const float* __restrict__ mu_rho,
                          const float* __restrict__ mu_theta,
                          const float* __restrict__ sigma_rho,
                          const float* __restrict__ sigma_theta,
                          const float* __restrict__ W_conv,
                          const float* __restrict__ b_conv,
                          float* __restrict__ out)
{
    const int bf  = blockIdx.x;          // 0 .. B*F-1
    const int b   = bf / F_;
    const int f   = bf - b * F_;
    const int tid = threadIdx.x;         // 0 .. 159

    __shared__ float s_rho[V_];
    __shared__ float s_th[V_];
    __shared__ float s_ft[V_];
    __shared__ float s_desc[NROT_ * DSTRIDE];   // desc[r][k], row M = rotation r

    if (tid < V_) {
        s_rho[tid] = rhos[b * V_ + tid];
        s_th[tid]  = thetas[b * V_ + tid];
        s_ft[tid]  = feature[(b * V_ + tid) * F_ + f];
    }
    __syncthreads();

    // ---------------- Phase 1: desc[r,k] into LDS ----------------
    // thread -> (k, rotation half). rho Gaussian computed once, reused for 8 r's.
    const int k     = tid % BINS_;            // 0..79
    const int rbase = (tid / BINS_) * 8;      // 0 or 8

    const float mrk = mu_rho   [f * BINS_ + k];
    const float mtk = mu_theta [f * BINS_ + k];
    const float sr  = sigma_rho[f * BINS_ + k];
    const float st  = sigma_theta[f * BINS_ + k];
    const float sri = 1.0f / (sr * sr + EPS_);
    const float sti = 1.0f / (st * st + EPS_);

    float s0[8], s1[8];
#pragma unroll
    for (int i = 0; i < 8; ++i) { s0[i] = 0.0f; s1[i] = 0.0f; }

    for (int v = 0; v < V_; ++v) {
        const float rv = s_rho[v];
        const float tv = s_th[v];
        const float fv = s_ft[v];
        const float dr  = rv - mrk;
        const float rg  = __expf(-dr * dr * sri);
        const float rgf = rg * fv;
#pragma unroll
        for (int i = 0; i < 8; ++i) {
            const int r = rbase + i;
            float tk = tv + (float)r * ROT_STEP;
            tk -= TWO_PI * floorf(tk * INV_TWO_PI);        // jnp.mod(.., 2pi)
            const float dt = tk - mtk;
            const float tg = __expf(-dt * dt * sti);
            s0[i] += rg  * tg;
            s1[i] += rgf * tg;
        }
    }
#pragma unroll
    for (int i = 0; i < 8; ++i)
        s_desc[(rbase + i) * DSTRIDE + k] = s1[i] / (s0[i] + EPS_);

    __syncthreads();

    // ---------------- Phase 2: WMMA GEMM + rotation-max + bias + relu ----------
    // wave w handles output-column tile kc = w (16 columns). C tile is 16 rot x 16 bins.
    const int wave = tid >> 5;               // 0..4
    const int lane = tid & 31;
    const int kc   = wave;
    const int half = lane >> 4;              // K lane-half select (ISA 16x16x4 layout)
    const int l16  = lane & 15;              // A: row M ; B: column N
    const float* __restrict__ Wf = W_conv + f * BINS_ * BINS_;

    v8f c = {};
#pragma unroll
    for (int k0 = 0; k0 < BINS_; k0 += 4) {
        const int ka = k0 + half * 2;
        v2f a, bm;
        // A (16x4 f32): lanes 0-15 hold K=k0,k0+1 ; lanes 16-31 hold K=k0+2,k0+3
        a.x  = s_desc[l16 * DSTRIDE + ka + 0];
        a.y  = s_desc[l16 * DSTRIDE + ka + 1];
        // B (4x16 f32): column N = l16, same K split across lane halves
        bm.x = Wf[(ka + 0) * BINS_ + kc * 16 + l16];
        bm.y = Wf[(ka + 1) * BINS_ + kc * 16 + l16];
        c = __builtin_amdgcn_wmma_f32_16x16x4_f32(
                /*neg_a=*/false, a, /*neg_b=*/false, bm,
                /*c_mod=*/(short)0, c, /*reuse_a=*/false, /*reuse_b=*/false);
    }

    // C layout: c[j] -> (M=j, N=l16) for lanes 0-15, (M=j+8, N=l16) for lanes 16-31.
    // Bias is rotation-invariant: max_r(conv) = max_r(desc@W) + bias.
    float m = c[0];
#pragma unroll
    for (int j = 1; j < 8; ++j) m = fmaxf(m, c[j]);
    const float mo = __shfl_xor(m, 16, 32);  // combine M=0..7 with M=8..15
    m = fmaxf(m, mo);

    if (half == 0) {
        const int n = kc * 16 + l16;
        const float val = m + b_conv[f * BINS_ + n];
        out[b * (F_ * BINS_) + f * BINS_ + n] = fmaxf(val, 0.0f);
    }
}

extern "C" void kernel_launch(void* const* d_in, const int* in_sizes, int n_in,
                              void* d_out, int out_size, void* d_ws, size_t ws_size,
                              hipStream_t stream) {
    (void)in_sizes; (void)n_in; (void)out_size; (void)d_ws; (void)ws_size;
    const float* feature     = (const float*)d_in[0];
    const float* rhos        = (const float*)d_in[1];
    const float* thetas      = (const float*)d_in[2];
    const float* mu_rho      = (const float*)d_in[3];
    const float* mu_theta    = (const float*)d_in[4];
    const float* sigma_rho   = (const float*)d_in[5];
    const float* sigma_theta = (const float*)d_in[6];
    const float* W_conv      = (const float*)d_in[7];
    const float* b_conv      = (const float*)d_in[8];
    float* out = (float*)d_out;

    dim3 grid(B_ * F_);      // 5120 blocks: one per (batch, feature)
    dim3 block(160);         // 5 waves: phase1 k x rot-half map, phase2 one wave per col tile
    GaussianFiler_kernel<<<grid, block, 0, stream>>>(
        feature, rhos, thetas, mu_rho, mu_theta, sigma_rho, sigma_theta,
        W_conv, b_conv, out);
}